// MmModel_2568390443041
// MI455X (gfx1250) — compile-verified
//
#include <hip/hip_runtime.h>
#include <hip/hip_bf16.h>
#include <stdint.h>

#define NU 50000
#define NI 50000
#define NN 100000
#define EMB 64
#define NLAYERS 3
#define NEDGES 1600000
#define BATCH 4096
#define CAT_RATE 0.02f

#ifndef __has_builtin
#define __has_builtin(x) 0
#endif

#if __has_builtin(__builtin_amdgcn_global_load_async_to_lds_b128)
#define HAVE_ASYNC 1
#else
#define HAVE_ASYNC 0
#endif

#if __has_builtin(__builtin_amdgcn_wmma_f32_16x16x4_f32)
#define HAVE_WMMA 1
#else
#define HAVE_WMMA 0
#endif

// 16-byte vector pointee type the async builtin expects (per compiler diagnostic):
//   param0: 'int __vector(4) __device__ *'  (AS1)
typedef int v4i __attribute__((vector_size(16)));
typedef __attribute__((address_space(1))) v4i gv4i;   // global (AS1)
typedef __attribute__((address_space(3))) v4i lv4i;   // LDS (AS3)

typedef float v2f __attribute__((ext_vector_type(2)));
typedef float v8f __attribute__((ext_vector_type(8)));

template <int CNT>
__device__ __forceinline__ void wait_async() {
#if __has_builtin(__builtin_amdgcn_s_wait_asynccnt)
    __builtin_amdgcn_s_wait_asynccnt(CNT);
#else
    asm volatile("s_wait_asynccnt %0" ::"i"(CNT) : "memory");
#endif
}

#if HAVE_ASYNC
__device__ __forceinline__ void async_copy_b128(const void* g, void* l) {
    // per-lane 16B: copies 16 bytes from the global address g to the LDS address l, tracked by ASYNCcnt
    __builtin_amdgcn_global_load_async_to_lds_b128(
        (gv4i*)(uintptr_t)g, (lv4i*)l, 0, 0);
}
#endif

// ---------------- init: cur_e = sum_e = E0 (vectorized) ----------------
__global__ __launch_bounds__(256) void k_init_copy(const float* __restrict__ E0,
                                                   float* __restrict__ cur_e,
                                                   float* __restrict__ sum_e) {
    size_t i = (size_t)(blockIdx.x * blockDim.x + threadIdx.x) * 4;
    float4 v = *(const float4*)(E0 + i);
    *(float4*)(cur_e + i) = v;
    *(float4*)(sum_e + i) = v;
}

// ------- transpose 64xN weight matrices into Nx64, init cur & sum -------
__global__ __launch_bounds__(256) void k_transpose_init(
    const float* __restrict__ Wimg, const float* __restrict__ Wtxt,
    float* __restrict__ cur_img, float* __restrict__ sum_img,
    float* __restrict__ cur_txt, float* __restrict__ sum_txt) {
    __shared__ float ti[64][65];
    __shared__ float tt[64][65];
    const int n0 = blockIdx.x * 64;
    const int t = threadIdx.x;
#pragma unroll
    for (int it = 0; it < 16; ++it) {
        int lin = it * 256 + t;
        int k = lin >> 6;
        int nn = lin & 63;
        int n = n0 + nn;
        float a = 0.f, b = 0.f;
        if (n < NN) {
            a = Wimg[(size_t)k * NN + n];
            b = Wtxt[(size_t)k * NN + n];
        }
        ti[k][nn] = a;
        tt[k][nn] = b;
    }
    __syncthreads();
#pragma unroll
    for (int it = 0; it < 16; ++it) {
        int lin = it * 256 + t;
        int nn = lin >> 6;
        int k = lin & 63;
        int n = n0 + nn;
        if (n < NN) {
            float a = ti[k][nn];
            float b = tt[k][nn];
            size_t o = (size_t)n * EMB + k;
            cur_img[o] = a; sum_img[o] = a;
            cur_txt[o] = b; sum_txt[o] = b;
        }
    }
}

// ------------- fused 3-matrix COO SpMM: out[row] += val * x[col] -------------
__global__ __launch_bounds__(256) void k_spmm(
    const int* __restrict__ rows, const int* __restrict__ cols,
    const float* __restrict__ vals,
    const float* __restrict__ xe, const float* __restrict__ xi, const float* __restrict__ xt,
    float* __restrict__ oe, float* __restrict__ oi, float* __restrict__ ot) {
    const int lane = threadIdx.x & 31;
#if HAVE_ASYNC
    // Per-wave double-buffered edge-metadata tiles staged via async-to-LDS.
    constexpr int TILE = 128;   // NEDGES % TILE == 0 (12500 tiles)
    constexpr int WAVES = 8;
    __shared__ int smeta[WAVES * 2 * 3 * TILE];
    const int wv = threadIdx.x >> 5;
    const int gw = blockIdx.x * WAVES + wv;
    const int nw = gridDim.x * WAVES;
    const int ntiles = NEDGES / TILE;
    int* const wbase = smeta + wv * (2 * 3 * TILE);

    auto issue = [&](int tile, int buf) {
        size_t eb = (size_t)tile * TILE;
        int* l0 = wbase + buf * (3 * TILE);
        async_copy_b128(rows + eb + lane * 4, l0 + lane * 4);
        async_copy_b128(cols + eb + lane * 4, l0 + TILE + lane * 4);
        async_copy_b128(vals + eb + lane * 4, l0 + 2 * TILE + lane * 4);
    };

    if (gw < ntiles) issue(gw, 0);
    int buf = 0;
    for (int tcur = gw; tcur < ntiles; tcur += nw) {
        int tnext = tcur + nw;
        if (tnext < ntiles) {
            issue(tnext, buf ^ 1);
            wait_async<3>();   // the 3 ops of the current buffer retired (in-order)
        } else {
            wait_async<0>();
        }
        const int* lr = wbase + buf * (3 * TILE);
        const int* lc = lr + TILE;
        const float* lv = (const float*)(lr + 2 * TILE);
        for (int j = 0; j < TILE; ++j) {
            if ((j & 15) == 0 && j + 16 < TILE) {
                // columns known a tile ahead -> prefetch gathered feature rows
                size_t cp = (size_t)lc[j + 16] * EMB + lane * 2;
                __builtin_prefetch(xe + cp, 0, 0);
                __builtin_prefetch(xi + cp, 0, 0);
                __builtin_prefetch(xt + cp, 0, 0);
            }
            int r = lr[j];              // LDS broadcast reads
            int c = lc[j];
            float v = lv[j];
            size_t ci = (size_t)c * EMB + lane * 2;
            size_t ri = (size_t)r * EMB + lane * 2;
            float2 ve = *(const float2*)(xe + ci);
            float2 vi = *(const float2*)(xi + ci);
            float2 vt = *(const float2*)(xt + ci);
            atomicAdd(oe + ri,     v * ve.x);
            atomicAdd(oe + ri + 1, v * ve.y);
            atomicAdd(oi + ri,     v * vi.x);
            atomicAdd(oi + ri + 1, v * vi.y);
            atomicAdd(ot + ri,     v * vt.x);
            atomicAdd(ot + ri + 1, v * vt.y);
        }
        buf ^= 1;
    }
#else
    // Fallback: coalesced per-lane metadata loads + wave broadcast via shuffles.
    const int wv = threadIdx.x >> 5;
    const int gw = blockIdx.x * (blockDim.x >> 5) + wv;
    const int nw = gridDim.x * (blockDim.x >> 5);
    for (int e0 = gw * 32; e0 < NEDGES; e0 += nw * 32) {
        int rr = rows[e0 + lane];
        int cc = cols[e0 + lane];
        float vv = vals[e0 + lane];
        for (int j = 0; j < 32; ++j) {
            int r = __shfl(rr, j);
            int c = __shfl(cc, j);
            float v = __shfl(vv, j);
            size_t ci = (size_t)c * EMB + lane * 2;
            size_t ri = (size_t)r * EMB + lane * 2;
            float2 ve = *(const float2*)(xe + ci);
            float2 vi = *(const float2*)(xi + ci);
            float2 vt = *(const float2*)(xt + ci);
            atomicAdd(oe + ri,     v * ve.x);
            atomicAdd(oe + ri + 1, v * ve.y);
            atomicAdd(oi + ri,     v * vi.x);
            atomicAdd(oi + ri + 1, v * vi.y);
            atomicAdd(ot + ri,     v * vt.x);
            atomicAdd(ot + ri + 1, v * vt.y);
        }
    }
#endif
}

// ---- per-layer epilogue: fold biases into nxt, accumulate layer sums ----
__global__ __launch_bounds__(256) void k_post(
    float* __restrict__ ne, float* __restrict__ nimg, float* __restrict__ ntx,
    const float* __restrict__ bi, const float* __restrict__ bt,
    float* __restrict__ se, float* __restrict__ simg, float* __restrict__ stx) {
    size_t i = (size_t)(blockIdx.x * blockDim.x + threadIdx.x) * 4;
    int k = (int)(i & 63);
    float4 e = *(const float4*)(ne + i);
    float4 a = *(const float4*)(nimg + i);
    float4 b = *(const float4*)(ntx + i);
    float4 vbi = *(const float4*)(bi + k);
    float4 vbt = *(const float4*)(bt + k);
    a.x += vbi.x; a.y += vbi.y; a.z += vbi.z; a.w += vbi.w;
    b.x += vbt.x; b.y += vbt.y; b.z += vbt.z; b.w += vbt.w;
    *(float4*)(nimg + i) = a;
    *(float4*)(ntx + i) = b;
    float4 s0 = *(const float4*)(se + i);
    float4 s1 = *(const float4*)(simg + i);
    float4 s2 = *(const float4*)(stx + i);
    s0.x += e.x; s0.y += e.y; s0.z += e.z; s0.w += e.w;
    s1.x += a.x; s1.y += a.y; s1.z += a.z; s1.w += a.w;
    s2.x += b.x; s2.y += b.y; s2.z += b.z; s2.w += b.w;
    *(float4*)(se + i) = s0;
    *(float4*)(simg + i) = s1;
    *(float4*)(stx + i) = s2;
}

// ---- final epilogue: one wave per 16 output rows.
// Row L2 norms = diagonal of X*X^T via V_WMMA_F32_16X16X4_F32 (16 K-steps).
// Layout coincidence (ISA 7.12.2, 32-bit A 16x4 and B 4x16 tables): for B = X^T,
// the B-operand holds exactly the same element in the same lane/VGPR slot as the
// A-operand holding X, so SRC0 == SRC1 computes the Gram matrix with no shuffles.
__global__ __launch_bounds__(256) void k_gather(
    const int* __restrict__ ui, const int* __restrict__ pii, const int* __restrict__ nii,
    const float* __restrict__ se, const float* __restrict__ simg, const float* __restrict__ stx,
    float* __restrict__ out) {
    const int lane = threadIdx.x & 31;
    const int wv = threadIdx.x >> 5;
    const int grp = blockIdx.x * 8 + wv;     // 0..767; 16 rows per wave-group
    const int s = grp >> 8;                  // 256 groups per index set
    const int gb = (grp & 255) * 16;         // base batch index within set
    const int r16 = lane & 15;
    const int hi = lane >> 4;                // 0: K%4 in {0,1}; 1: K%4 in {2,3}
    const int b = gb + r16;
    const int idx = (s == 0) ? ui[b] : (s == 1 ? pii[b] : nii[b]);
    const size_t rowbase = (size_t)((s == 0) ? idx : (NU + idx)) * EMB;

    float ri_s[16], rt_s[16];                // per-row normalize scales
#if HAVE_WMMA
    v8f di = {};                             // Gram accumulators (sum matrices)
    v8f dt = {};
#pragma unroll
    for (int k = 0; k < 16; ++k) {
        const int col = 4 * k + 2 * hi;
        float2 x2 = *(const float2*)(simg + rowbase + col);
        float2 y2 = *(const float2*)(stx + rowbase + col);
        v2f ai; ai.x = x2.x; ai.y = x2.y;    // serves as both A (X) and B (X^T)
        v2f at; at.x = y2.x; at.y = y2.y;
        di = __builtin_amdgcn_wmma_f32_16x16x4_f32(false, ai, false, ai,
                                                   (short)0, di, false, false);
        dt = __builtin_amdgcn_wmma_f32_16x16x4_f32(false, at, false, at,
                                                   (short)0, dt, false, false);
    }
    // diag(m): VGPR m%8, lane m (m<8) or lane m+16 (m>=8). mean = 0.25*sum.
#pragma unroll
    for (int m = 0; m < 16; ++m) {
        float vi_ = (m < 8) ? di[m] : di[m - 8];
        float vt_ = (m < 8) ? dt[m] : dt[m - 8];
        const int src = (m < 8) ? m : (m + 16);
        float g1 = __shfl(vi_, src);
        float g2 = __shfl(vt_, src);
        ri_s[m] = 1.f / fmaxf(0.25f * sqrtf(g1), 1e-12f);
        rt_s[m] = 1.f / fmaxf(0.25f * sqrtf(g2), 1e-12f);
    }
#else
    // Fallback: shuffle-reduction norms, same launch geometry.
#pragma unroll
    for (int m = 0; m < 16; ++m) {
        int idx_m = __shfl(idx, m);
        size_t row = (size_t)((s == 0) ? idx_m : (NU + idx_m)) * EMB + lane * 2;
        float2 xi2 = *(const float2*)(simg + row);
        float2 xt2 = *(const float2*)(stx + row);
        float p1 = xi2.x * xi2.x + xi2.y * xi2.y;
        float p2 = xt2.x * xt2.x + xt2.y * xt2.y;
#pragma unroll
        for (int o = 16; o > 0; o >>= 1) {
            p1 += __shfl_xor(p1, o);
            p2 += __shfl_xor(p2, o);
        }
        ri_s[m] = 1.f / fmaxf(0.25f * sqrtf(p1), 1e-12f);
        rt_s[m] = 1.f / fmaxf(0.25f * sqrtf(p2), 1e-12f);
    }
#endif

    const size_t C = (size_t)BATCH * EMB;
#pragma unroll
    for (int m = 0; m < 16; ++m) {
        int idx_m = __shfl(idx, m);
        size_t row = (size_t)((s == 0) ? idx_m : (NU + idx_m)) * EMB + lane * 2;
        float2 me = *(const float2*)(se + row);
        float2 mi = *(const float2*)(simg + row);
        float2 mt = *(const float2*)(stx + row);
        me.x *= 0.25f; me.y *= 0.25f;
        mi.x *= 0.25f; mi.y *= 0.25f;
        mt.x *= 0.25f; mt.y *= 0.25f;
        float2 oe;
        oe.x = me.x + CAT_RATE * mi.x * ri_s[m] + CAT_RATE * mt.x * rt_s[m];
        oe.y = me.y + CAT_RATE * mi.y * ri_s[m] + CAT_RATE * mt.y * rt_s[m];
        size_t ob = (size_t)(gb + m) * EMB + lane * 2;
        *(float2*)(out + (size_t)s * C + ob) = oe;        // ue / ie(pos) / ie(neg)
        *(float2*)(out + (size_t)(3 + s) * C + ob) = mi;  // uimg / iimg
        *(float2*)(out + (size_t)(6 + s) * C + ob) = mt;  // utxt / itxt
    }
}

extern "C" void kernel_launch(void* const* d_in, const int* in_sizes, int n_in,
                              void* d_out, int out_size, void* d_ws, size_t ws_size,
                              hipStream_t stream) {
    (void)in_sizes; (void)n_in; (void)out_size; (void)ws_size;
    const int*   ui   = (const int*)d_in[0];
    const int*   pii  = (const int*)d_in[1];
    const int*   nii  = (const int*)d_in[2];
    const int*   rows = (const int*)d_in[3];
    const int*   cols = (const int*)d_in[4];
    const float* vals = (const float*)d_in[5];
    const float* E0   = (const float*)d_in[6];
    const float* Wimg = (const float*)d_in[7];
    const float* Bimg = (const float*)d_in[8];
    const float* Wtxt = (const float*)d_in[9];
    const float* Btxt = (const float*)d_in[10];
    float* out = (float*)d_out;

    const size_t MSZ = (size_t)NN * EMB;   // 6.4M floats = 25.6 MB per matrix
    float* ws = (float*)d_ws;
    float* A = ws;             // ping buffers: e, img, txt (contiguous)
    float* B = ws + 3 * MSZ;   // pong buffers
    float* S = ws + 6 * MSZ;   // layer sums

    // layer-0 state and sum init
    k_init_copy<<<6250, 256, 0, stream>>>(E0, A, S);
    k_transpose_init<<<1563, 256, 0, stream>>>(Wimg, Wtxt,
                                               A + MSZ, S + MSZ,
                                               A + 2 * MSZ, S + 2 * MSZ);

    for (int layer = 0; layer < NLAYERS; ++layer) {
        float* cur = (layer & 1) ? B : A;
        float* nxt = (layer & 1) ? A : B;
        (void)hipMemsetAsync(nxt, 0, 3 * MSZ * sizeof(float), stream);
        k_spmm<<<1024, 256, 0, stream>>>(rows, cols, vals,
                                         cur, cur + MSZ, cur + 2 * MSZ,
                                         nxt, nxt + MSZ, nxt + 2 * MSZ);
        k_post<<<6250, 256, 0, stream>>>(nxt, nxt + MSZ, nxt + 2 * MSZ,
                                         Bimg, Btxt,
                                         S, S + MSZ, S + 2 * MSZ);
    }

    // 3*BATCH rows, 16 per wave, 8 waves per block -> 96 blocks
    k_gather<<<96, 256, 0, stream>>>(ui, pii, nii,
                                     S, S + MSZ, S + 2 * MSZ, out);
}